// DepthwiseSeparableConv_11879879540981
// MI455X (gfx1250) — compile-verified
//
#include <hip/hip_runtime.h>
#include <hip/hip_bf16.h>

// ---------------- problem constants ----------------
#define BB   64
#define CIN  128
#define COUT 256
#define HH   56
#define WW   56
#define HWP  (HH*WW)          // 3136
#define EPS  1e-5f

typedef __attribute__((ext_vector_type(2))) float v2f;
typedef __attribute__((ext_vector_type(8))) float v8f;
typedef __attribute__((ext_vector_type(4))) int   v4i;

static __device__ __forceinline__ v8f wmma_f32_k4(v2f a, v2f b, v8f c) {
    // D = A(16x4 f32) * B(4x16 f32) + C(16x16 f32)
    return __builtin_amdgcn_wmma_f32_16x16x4_f32(
        /*neg_a=*/false, a, /*neg_b=*/false, b,
        /*c_mod=*/(short)0, c, /*reuse_a=*/false, /*reuse_b=*/false);
}

// ---------------------------------------------------------------
// Kernel 1: pack pw_w [COUT][CIN] into WMMA A-fragment order.
// A-frag (16x4): lane<16 -> row m0+lane, k in {4ks, 4ks+1}
//                lane>=16 -> row m0+lane-16, k in {4ks+2, 4ks+3}
// wpk[((mt*32 + ks)*32 + lane)*2 + {0,1}]
// ---------------------------------------------------------------
__global__ void k_pack(const float* __restrict__ pw, float* __restrict__ wpk) {
    int t = blockIdx.x * 256 + threadIdx.x;       // 16*32*32 = 16384 threads
    if (t >= 16 * 32 * 32) return;
    int lane = t & 31;
    int ks   = (t >> 5) & 31;
    int mt   = t >> 10;
    int m = mt * 16 + (lane & 15);
    int k = ks * 4 + 2 * (lane >> 4);
    float2 v;
    v.x = pw[m * CIN + k];
    v.y = pw[m * CIN + k + 1];
    *(float2*)(wpk + (size_t)t * 2) = v;
}

// ---------------------------------------------------------------
// Kernel 2: LDS-tiled depthwise 3x3 + NHWC-reinterpret scramble.
// Block handles (batch b, 2 output rows, 64-channel half).
// Stage: x slab [4 rows][64 ch][56 w] -> LDS (coalesced, zero-padded
// halo rows). Compute from LDS (row stride 57 floats: gcd(57,64)=1
// -> conflict-free when lanes differ in channel). Writes of
// y[b][(h*56+w)*128 + c] are contiguous in c -> fully coalesced.
// ---------------------------------------------------------------
#define XPITCH 57

__global__ __launch_bounds__(256) void
k_dw(const float* __restrict__ x, const float* __restrict__ dww,
     float* __restrict__ y) {
    __shared__ float xt[4 * 64 * XPITCH];   // 58,368 B

    const int h0 = blockIdx.x * 2;          // 0..54 step 2
    const int c0 = blockIdx.y * 64;         // 0 or 64
    const int b  = blockIdx.z;

    // taps (tiny, broadcast via scalar cache)
    const float k00 = dww[0], k01 = dww[1], k02 = dww[2];
    const float k10 = dww[3], k11 = dww[4], k12 = dww[5];
    const float k20 = dww[6], k21 = dww[7], k22 = dww[8];

    // ---- stage x slab: rows h0-1 .. h0+2 (clamped to zero) ----
    for (int e = threadIdx.x; e < 4 * 64 * WW; e += 256) {
        int w    = e % WW;
        int rest = e / WW;
        int c    = rest & 63;
        int row  = rest >> 6;               // 0..3
        int g    = h0 - 1 + row;
        float v = 0.0f;
        if (g >= 0 && g < HH)
            v = x[((long long)b * CIN + (c0 + c)) * HWP + g * WW + w];
        xt[(row * 64 + c) * XPITCH + w] = v;
    }
    __syncthreads();

    // ---- compute: thread -> (c = t&63, w0 = t>>6), 2 rows x 14 w ----
    const int c  = threadIdx.x & 63;
    const int w0 = threadIdx.x >> 6;        // 0..3

    #pragma unroll
    for (int hh = 0; hh < 2; ++hh) {
        const float* r0 = &xt[((hh + 0) * 64 + c) * XPITCH];
        const float* r1 = &xt[((hh + 1) * 64 + c) * XPITCH];
        const float* r2 = &xt[((hh + 2) * 64 + c) * XPITCH];
        const long long ybase = (long long)b * CIN * HWP
                              + (long long)(h0 + hh) * WW * CIN + c0 + c;
        #pragma unroll
        for (int it = 0; it < 14; ++it) {
            int w = w0 + it * 4;
            float acc = k01 * r0[w] + k11 * r1[w] + k21 * r2[w];
            if (w > 0)
                acc += k00 * r0[w - 1] + k10 * r1[w - 1] + k20 * r2[w - 1];
            if (w < WW - 1)
                acc += k02 * r0[w + 1] + k12 * r1[w + 1] + k22 * r2[w + 1];
            y[ybase + (long long)w * CIN] = acc;
        }
    }
}

// ---------------------------------------------------------------
// Kernel 3: WMMA GEMM  out[b] = relu(W[256x128] * Y_b[128x3136])
// Block: 256 threads (8 waves), tile = 256(M) x 64(N), batch b.
// Y tile staged to LDS via CDNA5 async-LDS loads when available.
// Epilogue fuses ReLU + per-channel sum/sumsq (BN stats) atomics.
// ---------------------------------------------------------------
#define YSTR 72   // LDS row stride (floats): halves hit disjoint banks

#if defined(__has_builtin)
#if __has_builtin(__builtin_amdgcn_global_load_async_to_lds_b128)
#define HAVE_ASYNC_LDS 1
#endif
#endif

#ifdef HAVE_ASYNC_LDS
typedef __attribute__((address_space(1))) v4i* gv4i_p;   // global b128 src
typedef __attribute__((address_space(3))) v4i* lv4i_p;   // LDS b128 dst
#endif

__global__ __launch_bounds__(256) void
k_gemm(const float* __restrict__ y, const float* __restrict__ wpk,
       float* __restrict__ out, float* __restrict__ csum, float* __restrict__ csumsq) {
    __shared__ float Ylds[CIN * YSTR];

    const int b  = blockIdx.y;
    const int n0 = blockIdx.x * 64;
    const float* Yb = y + (long long)b * CIN * HWP;

    // Stage Y tile [128 x 64] into LDS (16B per lane per step).
    {
        int rsub = threadIdx.x >> 4;        // 0..15
        int col  = (threadIdx.x & 15) * 4;  // 0..60
#ifdef HAVE_ASYNC_LDS
        float* ybnc = const_cast<float*>(Yb);
        #pragma unroll
        for (int it = 0; it < 8; ++it) {
            int row = it * 16 + rsub;
            __builtin_amdgcn_global_load_async_to_lds_b128(
                (gv4i_p)(ybnc + (size_t)row * HWP + n0 + col),
                (lv4i_p)(&Ylds[row * YSTR + col]),
                0, 0);
        }
#if defined(__has_builtin) && __has_builtin(__builtin_amdgcn_s_wait_asynccnt)
        __builtin_amdgcn_s_wait_asynccnt(0);
#else
        asm volatile("s_wait_asynccnt 0x0" ::: "memory");
#endif
#else
        #pragma unroll
        for (int it = 0; it < 8; ++it) {
            int row = it * 16 + rsub;
            float4 v = *(const float4*)(Yb + (size_t)row * HWP + n0 + col);
            *(float4*)(&Ylds[row * YSTR + col]) = v;
        }
#endif
    }
    __syncthreads();

    const int lane = threadIdx.x & 31;
    const int wave = threadIdx.x >> 5;
    const int half = lane >> 4;     // 0: lanes 0-15, 1: lanes 16-31
    const int j    = lane & 15;

    v8f acc[2][4];
    #pragma unroll
    for (int i = 0; i < 2; ++i)
        #pragma unroll
        for (int jn = 0; jn < 4; ++jn)
            acc[i][jn] = (v8f)(0.0f);

    const int mt0 = wave * 2;
    // K loop: 32 steps of K=4
    for (int ks = 0; ks < 32; ++ks) {
        v2f a0 = *(const v2f*)(wpk + ((size_t)((mt0 + 0) * 32 + ks) * 32 + lane) * 2);
        v2f a1 = *(const v2f*)(wpk + ((size_t)((mt0 + 1) * 32 + ks) * 32 + lane) * 2);
        int k0 = ks * 4 + 2 * half;   // B-frag: lanes 0-15 hold K={0,1}, 16-31 K={2,3}
        #pragma unroll
        for (int jn = 0; jn < 4; ++jn) {
            v2f bb;
            bb.x = Ylds[(k0 + 0) * YSTR + jn * 16 + j];
            bb.y = Ylds[(k0 + 1) * YSTR + jn * 16 + j];
            acc[0][jn] = wmma_f32_k4(a0, bb, acc[0][jn]);
            acc[1][jn] = wmma_f32_k4(a1, bb, acc[1][jn]);
        }
    }

    // Epilogue: ReLU, store, fused BN statistics.
    // C-frag layout: VGPR r -> row mbase+r (lanes 0-15), row mbase+8+r (lanes 16-31).
    const long long outbase = ((long long)b * COUT) * HWP + n0;
    #pragma unroll
    for (int i = 0; i < 2; ++i) {
        int mbase = wave * 32 + i * 16;
        #pragma unroll
        for (int r = 0; r < 8; ++r) {
            int m = mbase + r + 8 * half;
            float s = 0.0f, q = 0.0f;
            #pragma unroll
            for (int jn = 0; jn < 4; ++jn) {
                float v = acc[i][jn][r];
                v = v > 0.0f ? v : 0.0f;
                out[outbase + (long long)m * HWP + jn * 16 + j] = v;
                s += v;
                q += v * v;
            }
            // reduce across the 16 lanes of this half-wave (wave32)
            #pragma unroll
            for (int off = 8; off > 0; off >>= 1) {
                s += __shfl_xor(s, off, 16);
                q += __shfl_xor(q, off, 16);
            }
            if (j == 0) {
                atomicAdd(&csum[m], s);
                atomicAdd(&csumsq[m], q);
            }
        }
    }
}

// ---------------------------------------------------------------
// Kernel 4: per-channel BN scale/shift from fused statistics.
// ---------------------------------------------------------------
__global__ void k_scale(const float* __restrict__ csum, const float* __restrict__ csumsq,
                        const float* __restrict__ gamma, const float* __restrict__ beta,
                        float* __restrict__ scale, float* __restrict__ shift) {
    int o = threadIdx.x;   // 256
    const float n = (float)((long long)BB * HWP);   // 200704
    float mean = csum[o] / n;
    float var  = csumsq[o] / n - mean * mean;
    float sc   = gamma[o] * rsqrtf(var + EPS);
    scale[o] = sc;
    shift[o] = beta[o] - mean * sc;
}

// ---------------------------------------------------------------
// Kernel 5: in-place normalize.
// ---------------------------------------------------------------
__global__ void k_bn(float* __restrict__ out, const float* __restrict__ scale,
                     const float* __restrict__ shift) {
    long long idx = (long long)blockIdx.x * blockDim.x + threadIdx.x;
    if (idx >= (long long)BB * COUT * HWP) return;
    int ch = (int)((idx / HWP) & (COUT - 1));
    out[idx] = out[idx] * scale[ch] + shift[ch];
}

// ---------------------------------------------------------------
extern "C" void kernel_launch(void* const* d_in, const int* in_sizes, int n_in,
                              void* d_out, int out_size, void* d_ws, size_t ws_size,
                              hipStream_t stream) {
    const float* x     = (const float*)d_in[0];   // [64,128,56,56]
    const float* dw_w  = (const float*)d_in[1];   // [1,1,3,3]
    const float* pw_w  = (const float*)d_in[2];   // [256,128]
    const float* gamma = (const float*)d_in[3];   // [256]
    const float* beta  = (const float*)d_in[4];   // [256]
    float* out = (float*)d_out;

    // workspace layout
    const size_t Y_BYTES   = (size_t)BB * CIN * HWP * sizeof(float);   // ~102.8 MB
    const size_t WPK_BYTES = (size_t)COUT * CIN * sizeof(float);       // 128 KB
    float* y     = (float*)d_ws;
    float* wpk   = (float*)((char*)d_ws + Y_BYTES);
    float* stats = (float*)((char*)d_ws + Y_BYTES + WPK_BYTES);
    float* csum   = stats;
    float* csumsq = stats + COUT;
    float* scale  = stats + 2 * COUT;
    float* shift  = stats + 3 * COUT;

    // zero BN accumulators (graph-capture safe)
    (void)hipMemsetAsync(stats, 0, 2 * COUT * sizeof(float), stream);

    // 1. pack pw_w into A-fragment layout
    k_pack<<<64, 256, 0, stream>>>(pw_w, wpk);

    // 2. depthwise conv + scramble (LDS-tiled stencil)
    {
        dim3 grid(HH / 2, CIN / 64, BB);   // 28 x 2 x 64
        k_dw<<<grid, 256, 0, stream>>>(x, dw_w, y);
    }

    // 3. WMMA GEMM + ReLU + fused BN stats
    {
        dim3 grid(HWP / 64, BB);   // 49 x 64
        k_gemm<<<grid, 256, 0, stream>>>(y, wpk, out, csum, csumsq);
    }

    // 4. stats -> scale/shift
    k_scale<<<1, COUT, 0, stream>>>(csum, csumsq, gamma, beta, scale, shift);

    // 5. normalize in place
    {
        long long total = (long long)BB * COUT * HWP;     // 51,380,224
        int blocks = (int)((total + 255) / 256);
        k_bn<<<blocks, 256, 0, stream>>>(out, scale, shift);
    }
}